// GATNet_53171695124571
// MI455X (gfx1250) — compile-verified
//
#include <hip/hip_runtime.h>
#include <math.h>

#define NNODES 50000
#define NEDGES 800000

typedef __attribute__((ext_vector_type(2))) float v2f;
typedef __attribute__((ext_vector_type(8))) float v8f;

// ---------------------------------------------------------------------------
// CDNA5 async Global->LDS copies (ASYNCcnt path) + wait
// ---------------------------------------------------------------------------
__device__ __forceinline__ void async_g2l_b32(unsigned lds_byte_off, const void* gaddr) {
  asm volatile("global_load_async_to_lds_b32 %0, %1, off"
               :: "v"(lds_byte_off), "v"(gaddr) : "memory");
}
__device__ __forceinline__ void wait_async0() {
#if __has_builtin(__builtin_amdgcn_s_wait_asynccnt)
  __builtin_amdgcn_s_wait_asynccnt(0);
#else
  asm volatile("s_wait_asynccnt 0x0" ::: "memory");
#endif
}

// ---------------------------------------------------------------------------
// float atomic max via int-ordering trick (init must be -inf)
// ---------------------------------------------------------------------------
__device__ __forceinline__ void atomicMaxF(float* addr, float v) {
  if (v >= 0.f)
    atomicMax((int*)addr, __float_as_int(v));
  else
    atomicMin((unsigned int*)addr, __float_as_uint(v));
}

// ---------------------------------------------------------------------------
// fp32 WMMA GEMM. Block = 8 waves = 128 rows; each wave 16 rows x HC cols.
// W is staged in KB=32 K-slabs through LDS by the async DMA engine,
// PAIR-INTERLEAVED: float2 at (pair p, col c) = {W[2p][c], W[2p+1][c]}.
// -> every WMMA B operand is one contiguous ds_load_b64 (no register movs).
// Pair-row stride S2 = HC+16 float2s => the two half-wave K offsets land
// 32 banks apart (mod 64): conflict-free.
// Double-buffered: DMA of slab s+1 overlaps WMMA on slab s.
// A 16x4 f32 layout: lane m=lane&15 holds row m; lanes>=16 hold K+2,K+3.
// C/D: VGPR r -> (M=r, N=lane) / (M=8+r, N=lane-16).
// ---------------------------------------------------------------------------
template <int D, int HC>
__global__ __launch_bounds__(256) void gemm_wmma_k(const float* __restrict__ X,
                                                   const float* __restrict__ W,
                                                   float* __restrict__ Hout,
                                                   int nrows) {
  constexpr int KB   = 32;                 // K slab (divides 32/64/128/256)
  constexpr int S2   = HC + 16;            // pair-row stride (float2 units)
  constexpr int SLAB = (KB / 2) * S2 * 2;  // floats per slab buffer
  constexpr int NT   = HC / 16;
  __shared__ float sW[2 * SLAB];

  const int tid  = (int)threadIdx.x;
  const int wib  = tid >> 5;               // wave in block (0..7)
  const int lane = tid & 31;
  const int row0 = (int)blockIdx.x * 128 + wib * 16;
  const bool valid = (row0 < nrows);
  const int rowb = valid ? row0 : 0;       // clamp so idle waves read safely

  const int m    = lane & 15;              // row (A) / col (B) within tile
  const int koff = (lane >> 4) * 2;        // lanes 16-31 handle K+2,K+3

  v8f acc[NT];
#pragma unroll
  for (int t = 0; t < NT; ++t) acc[t] = (v8f)(0.f);

  const float* xrow = X + (size_t)(rowb + m) * D;
  const unsigned ldsbase = (unsigned)(size_t)&sW[0];

  // stage one K-slab, pair-interleaving during the async copy
  auto stage = [&](int k0, int bufsel) {
    const unsigned base = ldsbase + (unsigned)(bufsel * SLAB) * 4u;
    constexpr int Q = KB * HC;             // b32 granules per slab
    for (int q = tid; q < Q; q += 256) {
      int kk = q / HC;                     // consecutive tid -> consecutive c
      int c  = q - kk * HC;
      const float* gp = W + (size_t)(k0 + kk) * HC + c;
      unsigned loff = base + (unsigned)(((kk >> 1) * S2 + c) * 2 + (kk & 1)) * 4u;
      async_g2l_b32(loff, gp);
    }
  };

  constexpr int NSLAB = D / KB;
  stage(0, 0);
  wait_async0();
  __syncthreads();

  for (int s = 0; s < NSLAB; ++s) {
    if (s + 1 < NSLAB) stage((s + 1) * KB, (s + 1) & 1);  // overlap DMA w/ WMMA

    const float* bufp = &sW[(s & 1) * SLAB];
    const int k0 = s * KB;
#pragma unroll
    for (int kk = 0; kk < KB; kk += 4) {
      v2f a;
      a.x = xrow[k0 + kk + koff];
      a.y = xrow[k0 + kk + koff + 1];
      const int p = (kk + koff) >> 1;                     // k-pair index
      const float* b0 = bufp + ((size_t)p * S2 + m) * 2;  // float2 base, col m
#pragma unroll
      for (int t = 0; t < NT; ++t) {
        v2f b = *reinterpret_cast<const v2f*>(b0 + t * 32);  // ds_load_b64
        acc[t] = __builtin_amdgcn_wmma_f32_16x16x4_f32(
            false, a, false, b, (short)0, acc[t], false, false);
      }
    }

    wait_async0();       // next slab's DMA complete (no-op on last iter)
    __syncthreads();     // all waves done reading current slab
  }

  if (valid) {
    const int n  = lane & 15;
    const int mb = (lane >> 4) * 8;
#pragma unroll
    for (int t = 0; t < NT; ++t) {
#pragma unroll
      for (int r = 0; r < 8; ++r)
        Hout[(size_t)(row0 + mb + r) * HC + t * 16 + n] = acc[t][r];
    }
  }
}

// Scalar GEMM for last layer (HC=3): negligible FLOPs.
__global__ void gemm_small_k(const float* __restrict__ X, const float* __restrict__ W,
                             float* __restrict__ Hout, int n, int D, int HC) {
  int i = blockIdx.x * blockDim.x + threadIdx.x;
  if (i >= n * HC) return;
  int row = i / HC, col = i - row * HC;
  const float* xr = X + (size_t)row * D;
  float s = 0.f;
  for (int k = 0; k < D; ++k) s += xr[k] * W[(size_t)k * HC + col];
  Hout[i] = s;
}

__global__ void fill_k(float* __restrict__ p, float v, long n) {
  long i = (long)blockIdx.x * blockDim.x + threadIdx.x;
  if (i < n) p[i] = v;
}

// es[n,h] = sum_c h[n,h,c]*a_src[h,c] ; ed likewise
__global__ void node_logits_k(const float* __restrict__ hf, const float* __restrict__ as_,
                              const float* __restrict__ ad_, float* __restrict__ es,
                              float* __restrict__ ed, int n, int H, int C) {
  int i = blockIdx.x * blockDim.x + threadIdx.x;
  if (i >= n * H) return;
  int node = i / H, h = i - node * H;
  const float* hp = hf + (size_t)node * H * C + (size_t)h * C;
  const float* ap = as_ + h * C;
  const float* dp = ad_ + h * C;
  float s = 0.f, d = 0.f;
  for (int c = 0; c < C; ++c) {
    float x = hp[c];
    s += x * ap[c];
    d += x * dp[c];
  }
  es[i] = s;
  ed[i] = d;
}

__device__ __forceinline__ void edge_sd(int e, const int* src, const int* dst,
                                        int& s, int& d) {
  if (e < NEDGES) { s = src[e]; d = dst[e]; }
  else            { s = e - NEDGES; d = s; }   // self-loop
}

__global__ void edge_max_k(const int* __restrict__ src, const int* __restrict__ dst,
                           const float* __restrict__ es, const float* __restrict__ ed,
                           float* __restrict__ emax, int H) {
  int t = blockIdx.x * blockDim.x + threadIdx.x;
  int Etot = NEDGES + NNODES;
  if (t >= Etot * H) return;
  int e = t / H, h = t - e * H;
  int s, d; edge_sd(e, src, dst, s, d);
  float v = es[s * H + h] + ed[d * H + h];
  v = (v > 0.f) ? v : 0.2f * v;                 // leaky_relu(0.2)
  atomicMaxF(&emax[d * H + h], v);
}

__global__ void edge_denom_k(const int* __restrict__ src, const int* __restrict__ dst,
                             const float* __restrict__ es, const float* __restrict__ ed,
                             const float* __restrict__ emax, float* __restrict__ denom,
                             int H) {
  int t = blockIdx.x * blockDim.x + threadIdx.x;
  int Etot = NEDGES + NNODES;
  if (t >= Etot * H) return;
  int e = t / H, h = t - e * H;
  int s, d; edge_sd(e, src, dst, s, d);
  float v = es[s * H + h] + ed[d * H + h];
  v = (v > 0.f) ? v : 0.2f * v;
  float p = __expf(v - emax[d * H + h]);
  atomicAdd(&denom[d * H + h], p);
}

// out[dst, hc] += h[src, hc] * alpha ; alpha recomputed in-register
template <int VEC>
__global__ void aggregate_k(const int* __restrict__ src, const int* __restrict__ dst,
                            const float* __restrict__ hf, const float* __restrict__ es,
                            const float* __restrict__ ed, const float* __restrict__ emax,
                            const float* __restrict__ denom, float* __restrict__ out,
                            int H, int C) {
  const int HC  = H * C;
  const int HCv = HC / VEC;
  long t = (long)blockIdx.x * blockDim.x + threadIdx.x;
  long total = (long)(NEDGES + NNODES) * HCv;
  if (t >= total) return;
  int e  = (int)(t / HCv);
  int c0 = (int)(t - (long)e * HCv) * VEC;      // absolute index inside HC
  int h  = c0 / C;
  int s, d; edge_sd(e, src, dst, s, d);
  float v = es[s * H + h] + ed[d * H + h];
  v = (v > 0.f) ? v : 0.2f * v;
  float alpha = __expf(v - emax[d * H + h]) / denom[d * H + h];
  const float* hp = hf + (size_t)s * HC + c0;
  float* op = out + (size_t)d * HC + c0;
  if (VEC == 4) {
    float4 hv = *reinterpret_cast<const float4*>(hp);
    atomicAdd(op + 0, hv.x * alpha);
    atomicAdd(op + 1, hv.y * alpha);
    atomicAdd(op + 2, hv.z * alpha);
    atomicAdd(op + 3, hv.w * alpha);
  } else {
#pragma unroll
    for (int j = 0; j < VEC; ++j) atomicAdd(op + j, hp[j] * alpha);
  }
}

// mid-layer epilogue: bias + leaky_relu(0.02), in place
__global__ void finalize_k(float* __restrict__ out, const float* __restrict__ bias,
                           long total, int HC) {
  long i = (long)blockIdx.x * blockDim.x + threadIdx.x;
  if (i >= total) return;
  float v = out[i] + bias[i % HC];
  out[i] = (v > 0.f) ? v : 0.02f * v;
}

// last-layer epilogue: bias + tanh()*0.5 -> d_out
__global__ void final_out_k(const float* __restrict__ acc, const float* __restrict__ bias,
                            float* __restrict__ out, long total, int HC) {
  long i = (long)blockIdx.x * blockDim.x + threadIdx.x;
  if (i >= total) return;
  float v = acc[i] + bias[i % HC];
  out[i] = 0.5f * tanhf(v);
}

// ---------------------------------------------------------------------------
// host orchestration (stream-ordered, graph-capture safe)
// ---------------------------------------------------------------------------
extern "C" void kernel_launch(void* const* d_in, const int* in_sizes, int n_in,
                              void* d_out, int out_size, void* d_ws, size_t ws_size,
                              hipStream_t stream) {
  (void)in_sizes; (void)n_in; (void)out_size; (void)ws_size;
  static const int HEADS[7] = {4, 4, 4, 4, 4, 4, 1};
  static const int CHS[7]   = {16, 32, 64, 64, 64, 32, 3};

  const float* x   = (const float*)d_in[0];
  const int*   ei  = (const int*)d_in[1];
  const int*   src = ei;            // edge_index[0]
  const int*   dst = ei + NEDGES;   // edge_index[1]

  float* bufA  = (float*)d_ws;                     // layer input / agg output
  float* bufB  = bufA + (size_t)NNODES * 256;      // transformed features h
  float* es    = bufB + (size_t)NNODES * 256;
  float* ed    = es + (size_t)NNODES * 4;
  float* emax  = ed + (size_t)NNODES * 4;
  float* denom = emax + (size_t)NNODES * 4;

  const int Etot = NEDGES + NNODES;
  const unsigned gemm_blk = (NNODES + 127) / 128;  // 128 rows per block

  for (int i = 0; i < 7; ++i) {
    const float* W  = (const float*)d_in[2 + 4 * i];
    const float* as = (const float*)d_in[3 + 4 * i];
    const float* ad = (const float*)d_in[4 + 4 * i];
    const float* bi = (const float*)d_in[5 + 4 * i];
    const int H = HEADS[i], C = CHS[i], HC = H * C;
    const float* Xin = (i == 0) ? x : bufA;

    // 1) h = Xin @ W  (fp32 WMMA; W staged via async Global->LDS DMA)
    switch (i) {
      case 0: gemm_wmma_k<32, 64>  <<<gemm_blk, 256, 0, stream>>>(Xin, W, bufB, NNODES); break;
      case 1: gemm_wmma_k<64, 128> <<<gemm_blk, 256, 0, stream>>>(Xin, W, bufB, NNODES); break;
      case 2: gemm_wmma_k<128, 256><<<gemm_blk, 256, 0, stream>>>(Xin, W, bufB, NNODES); break;
      case 3: gemm_wmma_k<256, 256><<<gemm_blk, 256, 0, stream>>>(Xin, W, bufB, NNODES); break;
      case 4: gemm_wmma_k<256, 256><<<gemm_blk, 256, 0, stream>>>(Xin, W, bufB, NNODES); break;
      case 5: gemm_wmma_k<256, 128><<<gemm_blk, 256, 0, stream>>>(Xin, W, bufB, NNODES); break;
      default:
        gemm_small_k<<<(NNODES * 3 + 255) / 256, 256, 0, stream>>>(Xin, W, bufB, NNODES, 128, 3);
        break;
    }

    // 2) per-node attention logits
    node_logits_k<<<(NNODES * H + 255) / 256, 256, 0, stream>>>(bufB, as, ad, es, ed,
                                                                NNODES, H, C);

    // 3) init segment state + output accumulator (Xin in bufA is dead now)
    fill_k<<<(NNODES * H + 255) / 256, 256, 0, stream>>>(emax, -INFINITY, (long)NNODES * H);
    fill_k<<<(NNODES * H + 255) / 256, 256, 0, stream>>>(denom, 0.f, (long)NNODES * H);
    fill_k<<<((long)NNODES * HC + 255) / 256, 256, 0, stream>>>(bufA, 0.f, (long)NNODES * HC);

    // 4) segment softmax over edges (max, then sum of exp)
    edge_max_k  <<<((long)Etot * H + 255) / 256, 256, 0, stream>>>(src, dst, es, ed, emax, H);
    edge_denom_k<<<((long)Etot * H + 255) / 256, 256, 0, stream>>>(src, dst, es, ed, emax, denom, H);

    // 5) weighted scatter-add aggregation (alpha recomputed in-register)
    if (C % 4 == 0) {
      long tot = (long)Etot * (HC / 4);
      aggregate_k<4><<<(unsigned)((tot + 255) / 256), 256, 0, stream>>>(
          src, dst, bufB, es, ed, emax, denom, bufA, H, C);
    } else {
      long tot = (long)Etot * HC;
      aggregate_k<1><<<(unsigned)((tot + 255) / 256), 256, 0, stream>>>(
          src, dst, bufB, es, ed, emax, denom, bufA, H, C);
    }

    // 6) epilogue
    if (i < 6) {
      finalize_k<<<((long)NNODES * HC + 255) / 256, 256, 0, stream>>>(
          bufA, bi, (long)NNODES * HC, HC);
    } else {
      final_out_k<<<((long)NNODES * 3 + 255) / 256, 256, 0, stream>>>(
          bufA, bi, (float*)d_out, (long)NNODES * 3, 3);
    }
  }
}